// SAGE_14448269984145
// MI455X (gfx1250) — compile-verified
//
#include <hip/hip_runtime.h>
#include <hip/hip_bf16.h>

typedef float v2f __attribute__((ext_vector_type(2)));
typedef float v8f __attribute__((ext_vector_type(8)));

constexpr int   NN       = 100000;
constexpr int   NE       = 1600000;
constexpr float BN_EPS   = 1e-5f;
constexpr float NORM_EPS = 1e-12f;

// ---------------------------------------------------------------- edge count
__global__ void k_count(const int* __restrict__ dst, float* __restrict__ cnt) {
  int e = blockIdx.x * blockDim.x + threadIdx.x;
  if (e < NE) atomicAdd(&cnt[dst[e]], 1.0f);
}

// ------------------------------------------------------- edge scatter (sum)
template <int DIM>
__global__ void k_scatter(const float* __restrict__ x, const int* __restrict__ src,
                          const int* __restrict__ dst, float* __restrict__ agg) {
  constexpr int C4 = DIM / 4;                 // float4 chunks per row
  long gid   = (long)blockIdx.x * blockDim.x + threadIdx.x;
  long total = (long)NE * C4;
  if (gid >= total) return;
  int e = (int)(gid / C4);
  int c = (int)(gid % C4);
  const float4 v = ((const float4*)(x + (long)src[e] * DIM))[c];
  float* a = agg + (long)dst[e] * DIM + c * 4;
  atomicAdd(a + 0, v.x);
  atomicAdd(a + 1, v.y);
  atomicAdd(a + 2, v.z);
  atomicAdd(a + 3, v.w);
}

// ---------------------------------------------------- fused SAGE GEMM (WMMA)
// out[16 rows x DOUT] per block.  FUSE2: out = (A1/max(cnt,1))@Wl + A2@Wr + b
// else: out = A1@Wl + b.  L2NORM: row-wise L2 normalization (torch F.normalize).
template <int DIN, int DOUT, bool FUSE2, bool L2NORM>
__global__ __launch_bounds__(32 * (DOUT / 16))
void k_sage_gemm(const float* __restrict__ A1, const float* __restrict__ cnt,
                 const float* __restrict__ A2, const float* __restrict__ Wl,
                 const float* __restrict__ Wr, const float* __restrict__ bias,
                 float* __restrict__ out) {
  constexpr int WAVES    = DOUT / 16;
  constexpr int NTHREADS = 32 * WAVES;

  __shared__ float lA1[16 * DIN];
  __shared__ float lA2[FUSE2 ? 16 * DIN : 1];
  __shared__ float lOut[16 * DOUT];
  __shared__ float lrow[16];

  const int tid  = threadIdx.x;
  const int lane = tid & 31;
  const int wave = tid >> 5;
  const int row0 = blockIdx.x * 16;

  if constexpr (FUSE2) {
    if (tid < 16) lrow[tid] = 1.0f / fmaxf(cnt[row0 + tid], 1.0f);
    __syncthreads();
  }
  // stage A tiles into LDS (coalesced)
  for (int idx = tid; idx < 16 * DIN; idx += NTHREADS) {
    int r = idx / DIN, c = idx % DIN;
    float a = A1[(long)(row0 + r) * DIN + c];
    if constexpr (FUSE2) {
      lA1[idx] = a * lrow[r];
      lA2[idx] = A2[(long)(row0 + r) * DIN + c];
    } else {
      lA1[idx] = a;
    }
  }
  __syncthreads();

  // V_WMMA_F32_16X16X4_F32 operand mapping (ISA 7.12.2):
  //   A: lane 0-15 -> M=lane, VGPR0/1 = K=k0,k0+1 ; lane 16-31 -> K=k0+2,k0+3
  //   B: lane -> N (within half), lane-half -> K msb, VGPR -> K lsb
  //   C/D: VGPR r -> M=r (lanes 0-15) / M=r+8 (lanes 16-31), N = lane&15
  const int m     = lane & 15;
  const int khalf = (lane >> 4) * 2;
  const int ncol  = wave * 16 + (lane & 15);

  v8f acc = {};
#pragma unroll
  for (int k0 = 0; k0 < DIN; k0 += 4) {
    v2f a, b;
    a.x = lA1[m * DIN + k0 + khalf];
    a.y = lA1[m * DIN + k0 + khalf + 1];
    b.x = Wl[(k0 + khalf) * DOUT + ncol];
    b.y = Wl[(k0 + khalf + 1) * DOUT + ncol];
    acc = __builtin_amdgcn_wmma_f32_16x16x4_f32(false, a, false, b, (short)0, acc,
                                                false, false);
  }
  if constexpr (FUSE2) {
#pragma unroll
    for (int k0 = 0; k0 < DIN; k0 += 4) {
      v2f a, b;
      a.x = lA2[m * DIN + k0 + khalf];
      a.y = lA2[m * DIN + k0 + khalf + 1];
      b.x = Wr[(k0 + khalf) * DOUT + ncol];
      b.y = Wr[(k0 + khalf + 1) * DOUT + ncol];
      acc = __builtin_amdgcn_wmma_f32_16x16x4_f32(false, a, false, b, (short)0, acc,
                                                  false, false);
    }
  }

  // spill accumulators + bias into LDS tile
  const float bval = bias[ncol];
#pragma unroll
  for (int r = 0; r < 8; ++r) {
    int mrow = r + 8 * (lane >> 4);
    lOut[mrow * DOUT + ncol] = acc[r] + bval;
  }
  __syncthreads();

  if constexpr (L2NORM) {
    if (tid < 16) {
      float s = 0.0f;
      for (int c = 0; c < DOUT; ++c) {
        float v = lOut[tid * DOUT + c];
        s += v * v;
      }
      lrow[tid] = 1.0f / fmaxf(sqrtf(s), NORM_EPS);
    }
    __syncthreads();
  }

  for (int idx = tid; idx < 16 * DOUT; idx += NTHREADS) {
    int r   = idx / DOUT;
    float v = lOut[idx];
    if constexpr (L2NORM) v *= lrow[r];
    out[(long)(row0 + r) * DOUT + (idx % DOUT)] = v;
  }
}

// ------------------------------------------------- batchnorm over relu(h1)
__global__ void k_bn_stats(const float* __restrict__ h, float* __restrict__ sums) {
  int ch  = threadIdx.x & 127;
  int sub = threadIdx.x >> 7;  // 0/1
  float s = 0.0f, s2 = 0.0f;
  for (int r = blockIdx.x * 2 + sub; r < NN; r += gridDim.x * 2) {
    float v = fmaxf(h[(long)r * 128 + ch], 0.0f);  // relu fused
    s += v;
    s2 += v * v;
  }
  atomicAdd(&sums[ch], s);
  atomicAdd(&sums[128 + ch], s2);
}

__global__ void k_bn_finalize(const float* __restrict__ sums,
                              const float* __restrict__ gamma,
                              const float* __restrict__ beta,
                              float* __restrict__ ss) {
  int c       = threadIdx.x;  // 128 threads
  float inv_n = 1.0f / (float)NN;
  float mean  = sums[c] * inv_n;
  float var   = sums[128 + c] * inv_n - mean * mean;  // biased, like torch BN
  float scale = gamma[c] * rsqrtf(var + BN_EPS);
  ss[c]       = scale;
  ss[128 + c] = beta[c] - mean * scale;
}

__global__ void k_bn_apply(const float* __restrict__ h, const float* __restrict__ ss,
                           float* __restrict__ o) {
  long i = (long)blockIdx.x * blockDim.x + threadIdx.x;
  if (i >= (long)NN * 128) return;
  int c = (int)(i & 127);
  float v = fmaxf(h[i], 0.0f);
  o[i] = v * ss[c] + ss[128 + c];
}

// --------------------------------------------------------------------- driver
extern "C" void kernel_launch(void* const* d_in, const int* in_sizes, int n_in,
                              void* d_out, int out_size, void* d_ws, size_t ws_size,
                              hipStream_t stream) {
  const float* x     = (const float*)d_in[0];
  const int*   eidx  = (const int*)d_in[1];
  const float* w1l   = (const float*)d_in[2];
  const float* b1l   = (const float*)d_in[3];
  const float* w1r   = (const float*)d_in[4];
  const float* gamma = (const float*)d_in[5];
  const float* beta  = (const float*)d_in[6];
  const float* w2l   = (const float*)d_in[7];
  const float* b2l   = (const float*)d_in[8];
  const float* w2r   = (const float*)d_in[9];
  const float* wfc   = (const float*)d_in[10];
  const float* bfc   = (const float*)d_in[11];
  const int* src = eidx;
  const int* dst = eidx + NE;

  float* w     = (float*)d_ws;
  float* cnt   = w;  w += NN;
  float* agg1  = w;  w += (long)NN * 64;
  float* h1    = w;  w += (long)NN * 128;
  float* hbn   = w;  w += (long)NN * 128;
  float* agg2  = w;  w += (long)NN * 128;
  float* h2    = w;  w += (long)NN * 128;
  float* bnsum = w;  w += 256;
  float* bnss  = w;  w += 256;

  hipMemsetAsync(cnt, 0, NN * sizeof(float), stream);
  hipMemsetAsync(agg1, 0, (size_t)NN * 64 * sizeof(float), stream);
  hipMemsetAsync(agg2, 0, (size_t)NN * 128 * sizeof(float), stream);
  hipMemsetAsync(bnsum, 0, 256 * sizeof(float), stream);

  k_count<<<(NE + 255) / 256, 256, 0, stream>>>(dst, cnt);

  {
    long total = (long)NE * 16;  // DIM=64 -> 16 float4 chunks
    k_scatter<64><<<(int)((total + 255) / 256), 256, 0, stream>>>(x, src, dst, agg1);
  }
  k_sage_gemm<64, 128, true, true><<<NN / 16, 256, 0, stream>>>(
      agg1, cnt, x, w1l, w1r, b1l, h1);

  k_bn_stats<<<1024, 256, 0, stream>>>(h1, bnsum);
  k_bn_finalize<<<1, 128, 0, stream>>>(bnsum, gamma, beta, bnss);
  {
    long total = (long)NN * 128;
    k_bn_apply<<<(int)((total + 255) / 256), 256, 0, stream>>>(h1, bnss, hbn);
  }

  {
    long total = (long)NE * 32;  // DIM=128 -> 32 float4 chunks
    k_scatter<128><<<(int)((total + 255) / 256), 256, 0, stream>>>(hbn, src, dst, agg2);
  }
  k_sage_gemm<128, 128, true, true><<<NN / 16, 256, 0, stream>>>(
      agg2, cnt, hbn, w2l, w2r, b2l, h2);

  // final FC: [N,128] @ [128,16] + bfc, no normalization — 1 wave per 16 rows
  k_sage_gemm<128, 16, false, false><<<NN / 16, 32, 0, stream>>>(
      h2, nullptr, nullptr, wfc, nullptr, bfc, (float*)d_out);
}